// RetinaNetLoss_86749749445341
// MI455X (gfx1250) — compile-verified
//
#include <hip/hip_runtime.h>

// ---- problem constants (match reference) ----
#define B_   8
#define A_   120000
#define G_   64
#define C_   81
#define PREDW (4 + C_)       // 85 floats per anchor row
#define ALPHA_ 0.25f

#define TPB  256             // 8 wave32 waves per block
#define APB  2048            // anchors per block (8 per thread)
#define GRIDX ((A_ + APB - 1) / APB)   // 59

typedef float v2f  __attribute__((ext_vector_type(2)));
typedef float v8f  __attribute__((ext_vector_type(8)));
typedef unsigned int u32x4 __attribute__((ext_vector_type(4)));
typedef int   i32x4 __attribute__((ext_vector_type(4)));
typedef int   i32x8 __attribute__((ext_vector_type(8)));

// ------------------------------------------------------------------
// TDM: 1-D tensor DMA of n_dw dwords from global -> LDS.
// Descriptor layout per CDNA5 ISA ch.8 (D# group0/group1).
// ------------------------------------------------------------------
__device__ __forceinline__ void tdm_load_1d(unsigned lds_off, const void* gsrc,
                                            unsigned n_dw) {
  unsigned long long ga = (unsigned long long)gsrc;
  u32x4 g0;
  g0[0] = 1u;                                        // count=1, user descriptor
  g0[1] = lds_off;                                   // lds_addr (bytes)
  g0[2] = (unsigned)(ga & 0xffffffffu);              // global_addr[31:0]
  g0[3] = (unsigned)((ga >> 32) & 0x01ffffffu)       // global_addr[56:32]
        | (2u << 30);                                // type = 2 ("image")
  i32x8 g1;
  g1[0] = (int)(2u << 16);                           // data_size = 2 (4 bytes)
  g1[1] = (int)((n_dw & 0xffffu) << 16);             // tensor_dim0[15:0]
  g1[2] = (int)(((n_dw >> 16) & 0xffffu)             // tensor_dim0[31:16]
        | (1u << 16));                               // tensor_dim1 = 1
  g1[3] = (int)((n_dw & 0xffffu) << 16);             // tile_dim0 = n_dw
  g1[4] = 1;                                         // tile_dim1=1, tile_dim2=0
  g1[5] = (int)n_dw;                                 // tensor_dim0_stride
  g1[6] = 0;
  g1[7] = 0;
  i32x4 z4 = {0, 0, 0, 0};
#if __clang_major__ >= 23
  i32x8 z8 = {0, 0, 0, 0, 0, 0, 0, 0};
  __builtin_amdgcn_tensor_load_to_lds(g0, g1, z4, z4, z8, 0);
#else
  __builtin_amdgcn_tensor_load_to_lds(g0, g1, z4, z4, 0);
#endif
}

// ------------------------------------------------------------------
// Exact-f32 wave32 sum via V_WMMA_F32_16X16X4_F32:
//   A[m,0]=v(lane m), A[m,2]=v(lane m+16), other K = 0; B = ones
//   => D[m,n] = v[m] + v[m+16]; summing the 8 D vgprs per lane gives
//   rows 0..7 (lanes 0..15) or rows 8..15 (lanes 16..31); one
//   shfl_xor(16) finishes the 32-lane total.
// ------------------------------------------------------------------
__device__ __forceinline__ float wave_sum_wmma(float v) {
  v2f a;  a.x = v;    a.y = 0.0f;
  v2f bb; bb.x = 1.0f; bb.y = 1.0f;
  v8f c = {};
  v8f d = __builtin_amdgcn_wmma_f32_16x16x4_f32(false, a, false, bb,
                                                (short)0, c, false, false);
  float s = d[0] + d[1] + d[2] + d[3] + d[4] + d[5] + d[6] + d[7];
  s += __shfl_xor(s, 16, 32);
  return s;
}

__device__ __forceinline__ float smooth_l1(float d) {
  float ad = fabsf(d);
  return ad < 1.0f ? 0.5f * d * d : ad - 0.5f;
}

__global__ __launch_bounds__(TPB) void retina_loss_main(
    const float* __restrict__ pred, const float* __restrict__ anchors,
    const float* __restrict__ gt_boxes, const int* __restrict__ gt_valid,
    float* __restrict__ ws) {
  __shared__ float  s_raw[G_ * 5];
  __shared__ int    s_vraw[G_];
  __shared__ float4 s_gbox[G_];    // xyxy (degenerate if invalid)
  __shared__ float  s_garea[G_];   // w*h  (1.0 if invalid)
  __shared__ float4 s_gm[G_];      // cxcywh for encode
  __shared__ float  s_gcls[G_];    // class + 1
  __shared__ float  s_wl[TPB / 32], s_wn[TPB / 32];

  const int tid = threadIdx.x;
  const int b   = blockIdx.y;
  const int a0  = blockIdx.x * APB;

  // --- async TDM stage of this batch's GT block into LDS ---
  if (tid == 0) {
    tdm_load_1d((unsigned)(unsigned long long)(void*)s_raw,
                gt_boxes + (long long)b * G_ * 5, G_ * 5);
    tdm_load_1d((unsigned)(unsigned long long)(void*)s_vraw,
                gt_valid + b * G_, G_);
    __builtin_amdgcn_s_wait_tensorcnt(0);
  }
  __syncthreads();

  // --- preprocess GT once per block ---
  // Invalid GT -> degenerate box (+BIG,+BIG,-BIG,-BIG): intersection clamps
  // to 0 so it can never reach IoU >= 0.5; if it wins the argmax at IoU 0 the
  // anchor is negative and the loss contribution is identical to the
  // reference's -1 masking.
  if (tid < G_) {
    float cx = s_raw[tid * 5 + 0], cy = s_raw[tid * 5 + 1];
    float w  = s_raw[tid * 5 + 2], h  = s_raw[tid * 5 + 3];
    bool  v  = s_vraw[tid] > 0;
    float4 bx;
    bx.x = v ? cx - 0.5f * w :  3.0e38f;
    bx.y = v ? cy - 0.5f * h :  3.0e38f;
    bx.z = v ? cx + 0.5f * w : -3.0e38f;
    bx.w = v ? cy + 0.5f * h : -3.0e38f;
    s_gbox[tid]  = bx;
    s_garea[tid] = v ? w * h : 1.0f;
    float4 m; m.x = cx; m.y = cy; m.z = w; m.w = h;
    s_gm[tid]   = m;
    s_gcls[tid] = s_raw[tid * 5 + 4] + 1.0f;
  }
  __syncthreads();

  float lossv = 0.0f, nv = 0.0f;
  const int aEnd = (a0 + APB < A_) ? (a0 + APB) : A_;

  for (int a = a0 + tid; a < aEnd; a += TPB) {
    const float4 av = ((const float4*)anchors)[a];
    const float acx = av.x, acy = av.y, aw = av.z, ah = av.w;
    const float ax0 = acx - 0.5f * aw, ay0 = acy - 0.5f * ah;
    const float ax1 = acx + 0.5f * aw, ay1 = acy + 0.5f * ah;
    const float aarea = aw * ah;

    const float* row = pred + ((long long)b * A_ + a) * PREDW;
    __builtin_prefetch(row, 0, 1);        // global_prefetch_b8: loc + low cls
    __builtin_prefetch(row + 64, 0, 1);   // rest of the 85-float row

    // argmax IoU over G without division: keep (inter, union) of the best and
    // compare a/b > c/d as a*d > c*b (all unions > 0).
    float binter = -1.0f, buni = 1.0f;
    int   bidx = 0;
#pragma unroll 8
    for (int g = 0; g < G_; ++g) {
      const float4 gb = s_gbox[g];        // one ds_load_b128 broadcast
      const float  ga = s_garea[g];
      float iw = fmaxf(fminf(ax1, gb.z) - fmaxf(ax0, gb.x), 0.0f);
      float ih = fmaxf(fminf(ay1, gb.w) - fmaxf(ay0, gb.y), 0.0f);
      float inter = iw * ih;
      float uni   = aarea + ga - inter;
      if (inter * buni > binter * uni) { binter = inter; buni = uni; bidx = g; }
    }

    const bool pos = (2.0f * binter >= buni);  // iou >= 0.5

    // focal-style class loss for every anchor (gather one conf float)
    int tc = pos ? (int)s_gcls[bidx] : 0;
    float pt = row[4 + tc];
    float om = 1.0f - pt;
    lossv += -ALPHA_ * om * om * logf(pt);

    // smooth-L1 loc loss only for positive anchors
    if (pos) {
      const float4 m = s_gm[bidx];
      float ex = (m.x - acx) / aw;
      float ey = (m.y - acy) / ah;
      float ew = logf(m.z) - logf(aw);
      float eh = logf(m.w) - logf(ah);
      lossv += smooth_l1(row[0] - ex) + smooth_l1(row[1] - ey) +
               smooth_l1(row[2] - ew) + smooth_l1(row[3] - eh);
      nv += 1.0f;
    }
  }

  // --- block reduction: WMMA wave sums, then 8-wave combine ---
  float wl = wave_sum_wmma(lossv);
  float wn = wave_sum_wmma(nv);
  if ((tid & 31) == 0) {
    s_wl[tid >> 5] = wl;
    s_wn[tid >> 5] = wn;
  }
  __syncthreads();
  if (tid == 0) {
    float L = 0.0f, N = 0.0f;
    for (int w = 0; w < TPB / 32; ++w) { L += s_wl[w]; N += s_wn[w]; }
    const int bf = blockIdx.y * gridDim.x + blockIdx.x;
    ws[bf * 2 + 0] = L;
    ws[bf * 2 + 1] = N;
  }
}

__global__ __launch_bounds__(TPB) void retina_loss_final(
    const float* __restrict__ ws, int nblocks, float* __restrict__ out) {
  __shared__ float sl[TPB], sn[TPB];
  float L = 0.0f, N = 0.0f;
  for (int i = threadIdx.x; i < nblocks; i += TPB) {
    L += ws[2 * i + 0];
    N += ws[2 * i + 1];
  }
  sl[threadIdx.x] = L;
  sn[threadIdx.x] = N;
  __syncthreads();
  for (int s = TPB / 2; s > 0; s >>= 1) {
    if (threadIdx.x < s) {
      sl[threadIdx.x] += sl[threadIdx.x + s];
      sn[threadIdx.x] += sn[threadIdx.x + s];
    }
    __syncthreads();
  }
  if (threadIdx.x == 0) out[0] = sl[0] / sn[0];
}

extern "C" void kernel_launch(void* const* d_in, const int* in_sizes, int n_in,
                              void* d_out, int out_size, void* d_ws,
                              size_t ws_size, hipStream_t stream) {
  const float* pred     = (const float*)d_in[0];  // (B, A, 85) f32
  const float* anchors  = (const float*)d_in[1];  // (A, 4)     f32
  const float* gt_boxes = (const float*)d_in[2];  // (B, G, 5)  f32
  const int*   gt_valid = (const int*)d_in[3];    // (B, G)     i32
  float* ws  = (float*)d_ws;                      // 2 floats per block
  float* out = (float*)d_out;                     // scalar f32

  dim3 grid(GRIDX, B_);
  retina_loss_main<<<grid, TPB, 0, stream>>>(pred, anchors, gt_boxes, gt_valid,
                                             ws);
  retina_loss_final<<<1, TPB, 0, stream>>>(ws, GRIDX * B_, out);
}